// CosSimConv1D_70489003262497
// MI455X (gfx1250) — compile-verified
//
#include <hip/hip_runtime.h>
#include <hip/hip_bf16.h>

typedef float v2f __attribute__((ext_vector_type(2)));
typedef float v8f __attribute__((ext_vector_type(8)));

// Problem sizes (fixed by the harness/reference)
constexpr int Bsz = 32, T = 4096, C = 128, U = 256, K3 = 384;
constexpr int M_TILE   = 64;            // output rows per workgroup
constexpr int XS_ROWS  = M_TILE + 2;    // 66 staged input rows (halo of 1 each side)
constexpr int XS_PITCH = C + 4;         // 132: A-fragment b64 reads bank-conflict free
constexpr int WK       = 64;            // K rows of W staged per chunk
constexpr int W2P      = 272;           // float2 entries per k-row-pair slab (+16 pad
                                        //  -> half-waves land on disjoint bank halves)

// ---------------------------------------------------------------------------
// Prep: per-unit weight norm (+q^2) and p^2, computed once into d_ws.
// ---------------------------------------------------------------------------
__global__ __launch_bounds__(256) void cossim_prep(
    const float* __restrict__ w, const float* __restrict__ p,
    const float* __restrict__ q, float* __restrict__ wsbuf)
{
    const int u = threadIdx.x;              // 256 threads == U
    const float q2 = q[0] * q[0];
    float s = 0.0f;
    #pragma unroll 8
    for (int k = 0; k < K3; ++k) {
        float v = w[(size_t)k * U + u];
        s = fmaf(v, v, s);
    }
    wsbuf[u] = sqrtf(fmaxf(s, 1e-12f)) + q2;   // w_norm[u] + q^2
    float pv = p[u];
    wsbuf[U + u] = pv * pv;                    // p[u]^2
}

// ---------------------------------------------------------------------------
// Main: fused stack3 + cosine-sim GEMM + pow/sign/bias epilogue.
// One block = 64 rows x 256 units. 8 waves: 4 (M) x 2 (N).
// W staged in LDS pre-swizzled into B-fragment order: slab kp = k/2 holds
// {W[2kp][n], W[2kp+1][n]} contiguously -> one ds_load_b64 per B fragment.
// ---------------------------------------------------------------------------
__global__ __launch_bounds__(256) void cossim_main(
    const float* __restrict__ x, const float* __restrict__ w,
    const float* __restrict__ bias, const float* __restrict__ q,
    const float* __restrict__ wn, const float* __restrict__ p2,
    float* __restrict__ out)
{
    __shared__ float xs[XS_ROWS * XS_PITCH];     // input tile (with halo rows)
    __shared__ float ws2[(WK / 2) * W2P * 2];    // W chunk, fragment-interleaved
    __shared__ float ssumh[XS_ROWS * 2];         // per-half-row sum of squares
    __shared__ float rxn[M_TILE];                // 1 / (l2norm + q^2) per output row

    const int tid  = threadIdx.x;
    const int lane = tid & 31;
    const int wave = tid >> 5;
    const int ln   = lane & 15;                  // 0..15 within half-wave
    const int lh   = lane >> 4;                  // half-wave select
    const int m_base = (wave >> 1) * 16;         // 0,16,32,48
    const int n_base = (wave & 1) * 128;         // 0 or 128

    const int g0   = blockIdx.x * M_TILE;        // first global row of this tile
    const int bidx = g0 / T;
    const int t0   = g0 % T;                     // M_TILE divides T -> one batch per tile
    const float q2 = q[0] * q[0];

    // ---- stage X tile: rows t0-1 .. t0+M_TILE (zeros outside the batch) ----
    for (int i = tid; i < XS_ROWS * (C / 4); i += 256) {
        const int r  = i >> 5;                   // C/4 == 32 float4 per row
        const int c4 = i & 31;
        const int t  = t0 + r - 1;
        float4 v = make_float4(0.f, 0.f, 0.f, 0.f);
        if (t >= 0 && t < T)
            v = *(const float4*)(x + ((size_t)bidx * T + t) * C + c4 * 4);
        *(float4*)(xs + r * XS_PITCH + c4 * 4) = v;
    }
    __syncthreads();

    // ---- row sum-of-squares (2 threads per row), then reciprocal x_norm ----
    if (tid < XS_ROWS * 2) {
        const int r = tid >> 1, h = tid & 1;
        const float* rp = xs + r * XS_PITCH + h * (C / 2);
        float s = 0.0f;
        #pragma unroll
        for (int j = 0; j < C / 8; ++j) {
            float4 v = *(const float4*)(rp + j * 4);
            s = fmaf(v.x, v.x, s); s = fmaf(v.y, v.y, s);
            s = fmaf(v.z, v.z, s); s = fmaf(v.w, v.w, s);
        }
        ssumh[tid] = s;
    }
    __syncthreads();
    if (tid < M_TILE) {
        const float s = ssumh[2 * tid] + ssumh[2 * tid + 1] +
                        ssumh[2 * tid + 2] + ssumh[2 * tid + 3] +
                        ssumh[2 * tid + 4] + ssumh[2 * tid + 5];
        rxn[tid] = 1.0f / (sqrtf(fmaxf(s, 1e-12f)) + q2);
    }

    // ---- accumulators: 8 x (16x16 f32) per wave ----
    v8f acc[8];
    #pragma unroll
    for (int i = 0; i < 8; ++i) acc[i] = v8f{};

    // ---- K loop: 3 taps x 2 chunks of 64 K-rows of W ----
    for (int tap = 0; tap < 3; ++tap) {
        for (int ch = 0; ch < 2; ++ch) {
            __syncthreads();                     // previous chunk fully consumed
            const int wrow0 = tap * C + ch * WK;
            // stage W chunk, interleaving row pairs into fragment order
            for (int i = tid; i < (WK / 2) * (U / 4); i += 256) {
                const int kp = i >> 6;           // 0..31 : k-row pair
                const int c4 = i & 63;           // float4 column
                const float* gsrc = w + (size_t)(wrow0 + 2 * kp) * U + c4 * 4;
                float4 a = *(const float4*)gsrc;
                float4 b = *(const float4*)(gsrc + U);
                float* dst = ws2 + (kp * W2P + c4 * 4) * 2;   // 16B aligned
                *(float4*)(dst)     = make_float4(a.x, b.x, a.y, b.y);
                *(float4*)(dst + 4) = make_float4(a.z, b.z, a.w, b.w);
            }
            __syncthreads();

            // prefetch next W chunk into cache (gfx1250 global_prefetch path)
            const int nxt = tap * 2 + ch + 1;
            if (nxt < 6 && tid < WK)
                __builtin_prefetch(w + (size_t)((nxt >> 1) * C + (nxt & 1) * WK + tid) * U, 0, 0);

            // tap tau shifts the A rows by tau within the haloed tile
            const float* arow = xs + (m_base + ln + tap) * XS_PITCH + ch * WK + lh * 2;
            #pragma unroll 4
            for (int kk = 0; kk < WK; kk += 4) {
                v2f a = *(const v2f*)(arow + kk);              // A frag: 16x4 f32
                // B fragment base for this k-step: slab kp = kk/2 + lh
                const float* bp = ws2 + (((kk >> 1) + lh) * W2P + n_base + ln) * 2;
                #pragma unroll
                for (int nt = 0; nt < 8; ++nt) {
                    v2f bb = *(const v2f*)(bp + nt * 32);      // one ds_load_b64
                    acc[nt] = __builtin_amdgcn_wmma_f32_16x16x4_f32(
                        false, a, false, bb, (short)0, acc[nt], false, false);
                }
            }
        }
    }

    // ---- epilogue: y = sign(raw)*(|raw|/(xn*wn)+eps)^(p^2) + b ----
    float xr[8];
    #pragma unroll
    for (int r = 0; r < 8; ++r) xr[r] = rxn[m_base + r + lh * 8];

    const size_t orow0 = (size_t)g0 * U;
    #pragma unroll
    for (int nt = 0; nt < 8; ++nt) {
        const int n   = n_base + nt * 16 + ln;
        const float rwn = 1.0f / wn[n];
        const float p2v = p2[n];
        const float bbv = bias[n];
        #pragma unroll
        for (int r = 0; r < 8; ++r) {
            const int m = m_base + r + lh * 8;   // C/D layout: VGPR r -> row r (+8 hi half)
            const float a  = acc[nt][r];
            const float ay = fabsf(a) * (xr[r] * rwn) + 1e-12f;
            // |y|^(p^2) = exp2(p^2 * log2(|y|)) via hardware v_log_f32 / v_exp_f32
            const float pw = __builtin_amdgcn_exp2f(p2v * __builtin_amdgcn_logf(ay));
            const float sg = (a > 0.f) ? 1.f : ((a < 0.f) ? -1.f : 0.f);
            out[orow0 + (size_t)m * U + n] = fmaf(sg, pw, bbv);
        }
    }
}

// ---------------------------------------------------------------------------
extern "C" void kernel_launch(void* const* d_in, const int* in_sizes, int n_in,
                              void* d_out, int out_size, void* d_ws, size_t ws_size,
                              hipStream_t stream) {
    (void)in_sizes; (void)n_in; (void)out_size; (void)ws_size;
    const float* x    = (const float*)d_in[0];   // [B,T,C]
    const float* w    = (const float*)d_in[1];   // [1,3C,U]
    const float* bias = (const float*)d_in[2];   // [U]
    const float* p    = (const float*)d_in[3];   // [U]
    const float* q    = (const float*)d_in[4];   // [1]
    float* out = (float*)d_out;                  // [B,T,U]
    float* wsf = (float*)d_ws;                   // [0..U): w_norm+q^2, [U..2U): p^2

    cossim_prep<<<1, 256, 0, stream>>>(w, p, q, wsf);
    cossim_main<<<(Bsz * T) / M_TILE, 256, 0, stream>>>(
        x, w, bias, q, wsf, wsf + U, out);
}